// MultiHeadDenseDotProductAttentionLayer_29085518528617
// MI455X (gfx1250) — compile-verified
//
#include <hip/hip_runtime.h>
#include <hip/hip_bf16.h>

typedef __attribute__((ext_vector_type(2))) float v2f;
typedef __attribute__((ext_vector_type(8))) float v8f;

#define NN   3072
#define KDIM 512
#define HH   8
#define DH   64
#define HD   512
#define ADIM 8
#define CLAMPV 5.0f

// ---------------------------------------------------------------------------
// CDNA5 async global->LDS copy (ASYNCcnt path), with safe fallback.
// Builtin signature (per hipcc diagnostic): (AS1 int*, AS3 int*, imm, imm)
// ---------------------------------------------------------------------------
#if defined(__gfx1250__) && __has_builtin(__builtin_amdgcn_global_load_async_to_lds_b32)
#define USE_ASYNC 1
#else
#define USE_ASYNC 0
#endif

typedef __attribute__((address_space(1))) int as1_int;
typedef __attribute__((address_space(3))) int as3_int;

__device__ __forceinline__ void cp_async_f32(float* l, const float* g) {
#if USE_ASYNC
    __builtin_amdgcn_global_load_async_to_lds_b32(
        (as1_int*)g, (as3_int*)l, 0, 0);
#else
    *l = *g;
#endif
}

__device__ __forceinline__ void wait_async() {
#if USE_ASYNC
#if __has_builtin(__builtin_amdgcn_s_wait_asynccnt)
    __builtin_amdgcn_s_wait_asynccnt(0);
#else
    asm volatile("s_wait_asynccnt 0x0" ::: "memory");
#endif
#endif
}

// ---------------------------------------------------------------------------
// K1: row softmax of S [8, 256]
// ---------------------------------------------------------------------------
__global__ void softmax_S_kernel(const float* __restrict__ S, float* __restrict__ Ssm) {
    __shared__ float red[256];
    int a = blockIdx.x;           // row 0..7
    int j = threadIdx.x;          // 0..255
    float v = S[a * 256 + j];
    red[j] = v;
    __syncthreads();
    for (int off = 128; off > 0; off >>= 1) {
        if (j < off) red[j] = fmaxf(red[j], red[j + off]);
        __syncthreads();
    }
    float mx = red[0];
    __syncthreads();
    float e = expf(v - mx);
    red[j] = e;
    __syncthreads();
    for (int off = 128; off > 0; off >>= 1) {
        if (j < off) red[j] += red[j + off];
        __syncthreads();
    }
    float s = red[0];
    Ssm[a * 256 + j] = e / s;
}

// ---------------------------------------------------------------------------
// K2: theta = angles @ Ssm ; cos/sin tables [N, 256]
// ---------------------------------------------------------------------------
__global__ void theta_kernel(const float* __restrict__ angles, const float* __restrict__ Ssm,
                             float* __restrict__ cosT, float* __restrict__ sinT) {
    __shared__ float ang[ADIM];
    int n = blockIdx.x;
    int j = threadIdx.x;  // 0..255
    if (j < ADIM) ang[j] = angles[n * ADIM + j];
    __syncthreads();
    float th = 0.0f;
#pragma unroll
    for (int a = 0; a < ADIM; ++a) th += ang[a] * Ssm[a * 256 + j];
    cosT[n * 256 + j] = cosf(th);
    sinT[n * 256 + j] = sinf(th);
}

// ---------------------------------------------------------------------------
// K3: fp32 WMMA GEMM: out = x @ W + b   (3072 x 512) @ (512 x 512)
// blockIdx.z selects Q/K/V. 64x64 tile per block, 4 waves, 16 rows per wave.
// Double-buffered async-staged LDS tiles.
// ---------------------------------------------------------------------------
__global__ void __launch_bounds__(128)
qkv_gemm_kernel(const float* __restrict__ x,
                const float* __restrict__ Wq, const float* __restrict__ bq,
                const float* __restrict__ Wk, const float* __restrict__ bk,
                const float* __restrict__ Wv, const float* __restrict__ bv,
                float* __restrict__ Qb, float* __restrict__ Kb, float* __restrict__ Vb) {
    __shared__ float ldsA[2][64 * 66];   // [row][k]  (+2 pad)
    __shared__ float ldsWT[2][64 * 66];  // [col][k]  (+2 pad)

    int sel = blockIdx.z;
    const float* W   = (sel == 0) ? Wq : (sel == 1) ? Wk : Wv;
    const float* bia = (sel == 0) ? bq : (sel == 1) ? bk : bv;
    float* outb      = (sel == 0) ? Qb : (sel == 1) ? Kb : Vb;

    int tid  = threadIdx.x;
    int wave = tid >> 5;
    int lane = tid & 31;
    int m16  = lane & 15;
    int half = lane >> 4;

    int blockRow = blockIdx.y * 64;
    int colBase  = blockIdx.x * 64;

    auto stage = [&](int buf, int kc) {
        // A tile [64 rows][64 k]  (global coalesced over k)
        for (int idx = tid; idx < 64 * 64; idx += 128) {
            int r = idx >> 6, k = idx & 63;
            cp_async_f32(&ldsA[buf][r * 66 + k],
                         &x[(size_t)(blockRow + r) * KDIM + kc + k]);
        }
        // W tile transposed -> [col][k]  (global coalesced over col)
        for (int idx = tid; idx < 64 * 64; idx += 128) {
            int k = idx >> 6, c = idx & 63;
            cp_async_f32(&ldsWT[buf][c * 66 + k],
                         &W[(size_t)(kc + k) * HD + colBase + c]);
        }
    };

    v8f acc[4];
#pragma unroll
    for (int t = 0; t < 4; ++t) acc[t] = v8f{0.f,0.f,0.f,0.f,0.f,0.f,0.f,0.f};

    stage(0, 0);
    int cur = 0;
    for (int kc = 0; kc < KDIM; kc += 64) {
        wait_async();
        __syncthreads();
        if (kc + 64 < KDIM) stage(cur ^ 1, kc + 64);

#pragma unroll
        for (int kk = 0; kk < 64; kk += 4) {
            v2f a2 = *(const v2f*)&ldsA[cur][(wave * 16 + m16) * 66 + kk + half * 2];
#pragma unroll
            for (int t = 0; t < 4; ++t) {
                v2f b2 = *(const v2f*)&ldsWT[cur][(t * 16 + m16) * 66 + kk + half * 2];
                acc[t] = __builtin_amdgcn_wmma_f32_16x16x4_f32(
                    false, a2, false, b2, (short)0, acc[t], false, false);
            }
        }
        cur ^= 1;
    }

    int wrow = blockRow + wave * 16;
#pragma unroll
    for (int t = 0; t < 4; ++t) {
        int col = colBase + t * 16 + m16;
        float bb = bia[col];
#pragma unroll
        for (int r = 0; r < 8; ++r) {
            outb[(size_t)(wrow + r + 8 * half) * HD + col] = acc[t][r] + bb;
        }
    }
}

// ---------------------------------------------------------------------------
// K4: in-place rotation of Q and K. One thread owns one (n,h) row of 64.
//   d < 32:  out[d] = q[d]*cos[d/2] - q[2d+1]*sin[d/2]
//   d >= 32: out[d] = q[d]*cos[d/2] + q[2(d-32)]*sin[d/2]
// ---------------------------------------------------------------------------
__global__ void rope_kernel(float* __restrict__ Qb, float* __restrict__ Kb,
                            const float* __restrict__ cosT, const float* __restrict__ sinT) {
    int gid = blockIdx.x * blockDim.x + threadIdx.x;
    int total = 2 * NN * HH;
    if (gid >= total) return;
    float* buf = (gid < NN * HH) ? Qb : Kb;
    int nh = gid % (NN * HH);
    int n = nh / HH, h = nh % HH;

    float* row = buf + (size_t)n * HD + h * DH;
    const float* cp = cosT + (size_t)n * 256 + h * 32;
    const float* sp = sinT + (size_t)n * 256 + h * 32;

    float q[64], cs[32], sn[32], o[64];
#pragma unroll
    for (int i = 0; i < 16; ++i) {
        float4 t4 = ((const float4*)row)[i];
        q[4 * i + 0] = t4.x; q[4 * i + 1] = t4.y; q[4 * i + 2] = t4.z; q[4 * i + 3] = t4.w;
    }
#pragma unroll
    for (int i = 0; i < 8; ++i) {
        float4 c4 = ((const float4*)cp)[i];
        float4 s4 = ((const float4*)sp)[i];
        cs[4 * i + 0] = c4.x; cs[4 * i + 1] = c4.y; cs[4 * i + 2] = c4.z; cs[4 * i + 3] = c4.w;
        sn[4 * i + 0] = s4.x; sn[4 * i + 1] = s4.y; sn[4 * i + 2] = s4.z; sn[4 * i + 3] = s4.w;
    }
#pragma unroll
    for (int d = 0; d < 32; ++d)
        o[d] = q[d] * cs[d >> 1] - q[2 * d + 1] * sn[d >> 1];
#pragma unroll
    for (int d = 32; d < 64; ++d)
        o[d] = q[d] * cs[d >> 1] + q[2 * (d - 32)] * sn[d >> 1];
#pragma unroll
    for (int i = 0; i < 16; ++i) {
        float4 t4; t4.x = o[4 * i]; t4.y = o[4 * i + 1]; t4.z = o[4 * i + 2]; t4.w = o[4 * i + 3];
        ((float4*)row)[i] = t4;
    }
}

// ---------------------------------------------------------------------------
// K5: flash attention per (row-tile of 64, head), double-buffered async tiles.
//  dest[h] = contiguous [N,64] slab at Qb + h*N*64 (raw reshape semantics)
//  src[h]  = contiguous [64,N] slab at Kb + h*64*N
//  V uses the true transpose: V[(m)*512 + h*64 + d]
// ---------------------------------------------------------------------------
__device__ __forceinline__ float half_reduce_max(float v) {
#pragma unroll
    for (int off = 1; off < 16; off <<= 1) v = fmaxf(v, __shfl_xor(v, off, 32));
    return v;
}
__device__ __forceinline__ float half_reduce_sum(float v) {
#pragma unroll
    for (int off = 1; off < 16; off <<= 1) v += __shfl_xor(v, off, 32);
    return v;
}

__global__ void __launch_bounds__(128)
attention_kernel(const float* __restrict__ Qb, const float* __restrict__ Kb,
                 const float* __restrict__ Vb, float* __restrict__ out) {
    __shared__ float ldsKT[2][64 * 66];    // [m][d] (+2 pad)  K^T tile
    __shared__ float ldsVT[2][64 * 66];    // [d][m] (+2 pad)  V^T tile
    __shared__ float ldsP[4 * 16 * 64];    // per-wave prob tile [16 rows][64 cols]

    int tid  = threadIdx.x;
    int wave = tid >> 5;
    int lane = tid & 31;
    int m16  = lane & 15;
    int half = lane >> 4;

    int h = blockIdx.y;
    int rowBase = blockIdx.x * 64 + wave * 16;

    const float* Qd = Qb + (size_t)h * NN * DH;   // [N,64] row-major
    const float* Ks = Kb + (size_t)h * DH * NN;   // [64,N] row-major

    auto stage = [&](int buf, int cBase) {
        // K tile -> ldsKT[m][d] : global coalesced over m, transposing scatter
        for (int idx = tid; idx < 64 * 64; idx += 128) {
            int d = idx >> 6, m = idx & 63;
            cp_async_f32(&ldsKT[buf][m * 66 + d], &Ks[(size_t)d * NN + cBase + m]);
        }
        // V tile -> ldsVT[d][m] : global coalesced over d, transposing scatter
        for (int idx = tid; idx < 64 * 64; idx += 128) {
            int m = idx >> 6, d = idx & 63;
            cp_async_f32(&ldsVT[buf][d * 66 + m],
                         &Vb[(size_t)(cBase + m) * HD + h * DH + d]);
        }
    };

    // Preload this wave's Q A-fragments (16 rows x 64 K) into registers.
    v2f qa[16];
    {
        const float* qrow = Qd + (size_t)(rowBase + m16) * DH;
#pragma unroll
        for (int i = 0; i < 16; ++i)
            qa[i] = *(const v2f*)(qrow + i * 4 + half * 2);
    }

    v8f acc[4];
#pragma unroll
    for (int t = 0; t < 4; ++t) acc[t] = v8f{0.f,0.f,0.f,0.f,0.f,0.f,0.f,0.f};
    float mrow[8], lrow[8];
#pragma unroll
    for (int r = 0; r < 8; ++r) { mrow[r] = -3.0e38f; lrow[r] = 0.0f; }

    const int pbase = wave * 16 * 64;

    stage(0, 0);
    int cur = 0;
    for (int ct = 0; ct < NN / 64; ++ct) {
        wait_async();
        __syncthreads();
        if (ct + 1 < NN / 64) stage(cur ^ 1, (ct + 1) * 64);

        // scores: 4 tiles of 16x16, K = 64 -> 16 wmma steps each
        v8f sc[4];
#pragma unroll
        for (int t = 0; t < 4; ++t) {
            v8f s = v8f{0.f,0.f,0.f,0.f,0.f,0.f,0.f,0.f};
#pragma unroll
            for (int kk = 0; kk < 16; ++kk) {
                v2f b2 = *(const v2f*)&ldsKT[cur][(t * 16 + m16) * 66 + kk * 4 + half * 2];
                s = __builtin_amdgcn_wmma_f32_16x16x4_f32(
                    false, qa[kk], false, b2, (short)0, s, false, false);
            }
#pragma unroll
            for (int r = 0; r < 8; ++r) {
                float v = s[r] * 0.125f;               // 1/sqrt(64)
                s[r] = fminf(fmaxf(v, -CLAMPV), CLAMPV);
            }
            sc[t] = s;
        }

        // online softmax update (rows live at r + 8*half within each lane-half)
#pragma unroll
        for (int r = 0; r < 8; ++r) {
            float tm = fmaxf(fmaxf(sc[0][r], sc[1][r]), fmaxf(sc[2][r], sc[3][r]));
            tm = half_reduce_max(tm);
            float mnew = fmaxf(mrow[r], tm);
            float scale = __expf(mrow[r] - mnew);
            mrow[r] = mnew;
            lrow[r] *= scale;
#pragma unroll
            for (int t = 0; t < 4; ++t) acc[t][r] *= scale;
            float su = 0.0f;
#pragma unroll
            for (int t = 0; t < 4; ++t) {
                float p = __expf(sc[t][r] - mnew);
                sc[t][r] = p;
                su += p;
            }
            lrow[r] += half_reduce_sum(su);
        }

        // round-trip probabilities through LDS to re-fragment as A operand
#pragma unroll
        for (int t = 0; t < 4; ++t)
#pragma unroll
            for (int r = 0; r < 8; ++r)
                ldsP[pbase + (r + 8 * half) * 64 + t * 16 + m16] = sc[t][r];

        // acc += P(16x64) @ V(64x64)
#pragma unroll
        for (int kk = 0; kk < 16; ++kk) {
            v2f pa = *(const v2f*)&ldsP[pbase + m16 * 64 + kk * 4 + half * 2];
#pragma unroll
            for (int t = 0; t < 4; ++t) {
                v2f vb = *(const v2f*)&ldsVT[cur][(t * 16 + m16) * 66 + kk * 4 + half * 2];
                acc[t] = __builtin_amdgcn_wmma_f32_16x16x4_f32(
                    false, pa, false, vb, (short)0, acc[t], false, false);
            }
        }
        cur ^= 1;
    }

    // normalize and write out[n][h][d]
#pragma unroll
    for (int t = 0; t < 4; ++t)
#pragma unroll
        for (int r = 0; r < 8; ++r)
            out[(size_t)(rowBase + r + 8 * half) * HD + h * DH + t * 16 + m16] =
                acc[t][r] / lrow[r];
}

// ---------------------------------------------------------------------------
extern "C" void kernel_launch(void* const* d_in, const int* in_sizes, int n_in,
                              void* d_out, int out_size, void* d_ws, size_t ws_size,
                              hipStream_t stream) {
    const float* x      = (const float*)d_in[0];
    const float* angles = (const float*)d_in[1];
    const float* Wq     = (const float*)d_in[2];
    const float* bq     = (const float*)d_in[3];
    const float* Wk     = (const float*)d_in[4];
    const float* bk     = (const float*)d_in[5];
    const float* Wv     = (const float*)d_in[6];
    const float* bv     = (const float*)d_in[7];
    const float* S      = (const float*)d_in[8];
    float* out = (float*)d_out;

    float* w = (float*)d_ws;
    float* Qb   = w;                              // N*512
    float* Kb   = Qb + (size_t)NN * HD;           // N*512
    float* Vb   = Kb + (size_t)NN * HD;           // N*512
    float* cosT = Vb + (size_t)NN * HD;           // N*256
    float* sinT = cosT + (size_t)NN * 256;        // N*256
    float* Ssm  = sinT + (size_t)NN * 256;        // 8*256

    softmax_S_kernel<<<ADIM, 256, 0, stream>>>(S, Ssm);
    theta_kernel<<<NN, 256, 0, stream>>>(angles, Ssm, cosT, sinT);
    qkv_gemm_kernel<<<dim3(HD / 64, NN / 64, 3), 128, 0, stream>>>(
        x, Wq, bq, Wk, bk, Wv, bv, Qb, Kb, Vb);
    {
        int total = 2 * NN * HH;
        rope_kernel<<<(total + 255) / 256, 256, 0, stream>>>(Qb, Kb, cosT, sinT);
    }
    attention_kernel<<<dim3(NN / 64, HH), 128, 0, stream>>>(Qb, Kb, Vb, out);
}